// MyEmbedding_18219251270279
// MI455X (gfx1250) — compile-verified
//
#include <hip/hip_runtime.h>

// Problem constants (match the reference)
#define VOCAB   32000
#define NSTATIC (VOCAB + 2)   // rows of weight + figure_token_weight
#define DDIM    5120
#define VDIM    768
#define BB      2
#define TT      512
#define TVTOK   32

typedef __attribute__((ext_vector_type(2))) float v2f;
typedef __attribute__((ext_vector_type(8))) float v8f;

// -----------------------------------------------------------------------------
// Kernel 1: vis[b,t,d] = sum_v vision_feat[b,t,v] * fc_w[d,v] + fc_b[d]
// One wave32 per 16x16 output tile, V_WMMA_F32_16X16X4_F32 along K=VDIM.
//   grid = (DDIM/16, TVTOK/16, BB), block = 32
// -----------------------------------------------------------------------------
__global__ __launch_bounds__(32)
void vis_proj_wmma_kernel(const float* __restrict__ vision_feat, // (B, TV, VD)
                          const float* __restrict__ fc_w,        // (D, VD)
                          const float* __restrict__ fc_b,        // (D,)
                          float* __restrict__ vis)               // (B, TV, D) in ws
{
    const int lane = threadIdx.x;        // 0..31
    const int half = lane >> 4;          // 0 or 1
    const int l    = lane & 15;

    const int nbase = blockIdx.x * 16;   // column tile in D
    const int mbase = blockIdx.y * 16;   // row tile in TV
    const int b     = blockIdx.z;

    // A: row m = mbase + l, contiguous K pair at k + 2*half
    const float* __restrict__ arow =
        vision_feat + ((size_t)b * TVTOK + (size_t)(mbase + l)) * VDIM;
    // B: column n = nbase + l of the GEMM == row n of fc_w (row-major (D,VD))
    const float* __restrict__ brow =
        fc_w + (size_t)(nbase + l) * VDIM;

    const int koff = half * 2;

    v8f c = {};
#pragma unroll 4
    for (int k = 0; k < VDIM; k += 4) {
        v2f a = *(const v2f*)(arow + k + koff);
        v2f bv = *(const v2f*)(brow + k + koff);
        // D = A*B + C  (f32, 16x16x4). 8 args:
        // (neg_a, A, neg_b, B, c_mod, C, reuse_a, reuse_b)
        c = __builtin_amdgcn_wmma_f32_16x16x4_f32(
                false, a, false, bv, (short)0, c, false, false);
    }

    const int n = nbase + l;
    const float bias = fc_b[n];
    float* __restrict__ visb = vis + (size_t)b * TVTOK * DDIM;
#pragma unroll
    for (int r = 0; r < 8; ++r) {
        const int m = mbase + r + half * 8;   // C/D layout: VGPR r -> M = r + 8*half
        visb[(size_t)m * DDIM + n] = c[r] + bias;
    }
}

// -----------------------------------------------------------------------------
// Kernel 2: per-token row gather. One 256-thread block per token, float4 copies.
//   grid = B*T, block = 256
// -----------------------------------------------------------------------------
__global__ __launch_bounds__(256)
void embed_gather_kernel(const int*   __restrict__ text,    // (B*T,)
                         const float* __restrict__ weight,  // (V, D)
                         const float* __restrict__ figure,  // (2, D)
                         const float* __restrict__ vis,     // (B, TV, D) in ws
                         float*       __restrict__ out)     // (B*T, D)
{
    const int token = blockIdx.x;            // 0 .. B*T-1
    const int b     = token / TT;

    int id = text[token];

    const float* __restrict__ src;
    if (id < NSTATIC) {
        int cid = id < 0 ? 0 : id;           // clip(text, 0, n_static-1)
        if (cid < VOCAB) src = weight + (size_t)cid * DDIM;
        else             src = figure + (size_t)(cid - VOCAB) * DDIM;
    } else {
        int vi = id - NSTATIC;               // clip to [0, TV-1]
        if (vi < 0)       vi = 0;
        if (vi >= TVTOK)  vi = TVTOK - 1;
        src = vis + ((size_t)b * TVTOK + (size_t)vi) * DDIM;
    }

    float* __restrict__ dst = out + (size_t)token * DDIM;

    const float4* __restrict__ s4 = (const float4*)src;
    float4*       __restrict__ d4 = (float4*)dst;
    const int n4 = DDIM / 4;                 // 1280
#pragma unroll 5
    for (int i = threadIdx.x; i < n4; i += 256) {
        d4[i] = s4[i];
    }
}

// -----------------------------------------------------------------------------
// Launch
// d_in order: text_input, vision_feat, weight, figure_token_weight, fc_w, fc_b
// -----------------------------------------------------------------------------
extern "C" void kernel_launch(void* const* d_in, const int* in_sizes, int n_in,
                              void* d_out, int out_size, void* d_ws, size_t ws_size,
                              hipStream_t stream) {
    const int*   text        = (const int*)  d_in[0];
    const float* vision_feat = (const float*)d_in[1];
    const float* weight      = (const float*)d_in[2];
    const float* figure      = (const float*)d_in[3];
    const float* fc_w        = (const float*)d_in[4];
    const float* fc_b        = (const float*)d_in[5];
    float*       out         = (float*)d_out;
    float*       vis         = (float*)d_ws;   // B*TV*D floats = 1.31 MB

    dim3 gridA(DDIM / 16, TVTOK / 16, BB);     // 320 x 2 x 2 tiles
    vis_proj_wmma_kernel<<<gridA, 32, 0, stream>>>(vision_feat, fc_w, fc_b, vis);

    dim3 gridB(BB * TT);                       // 1024 tokens
    embed_gather_kernel<<<gridB, 256, 0, stream>>>(text, weight, figure, vis, out);
}